// CrossAttentionSameFrame_42408507080959
// MI455X (gfx1250) — compile-verified
//
#include <hip/hip_runtime.h>

typedef __attribute__((ext_vector_type(2))) float v2f;
typedef __attribute__((ext_vector_type(4))) float v4f;
typedef __attribute__((ext_vector_type(8))) float v8f;

// ---------------------------------------------------------------------------
// One wave (32 lanes) computes one 16x16 tile of C = A(MxK) @ B(KxN) + bias,
// using V_WMMA_F32_16X16X4_F32 (full fp32 precision; compute cost negligible
// for this problem, so no reason to downcast).
//
// Per the CDNA5 ISA 32-bit WMMA layouts:
//   A 16x4 : lane m=lane&15, lanes 0-15 hold K={0,1} in v[0],v[1];
//            lanes 16-31 hold K={2,3}.
//   B 4x16 : lane n=lane&15, same K split across half-waves.
//   C/D    : vgpr i, lanes 0-15 -> M=i, lanes 16-31 -> M=i+8; N=lane&15.
// ---------------------------------------------------------------------------
__global__ __launch_bounds__(32)
void gemm16_wmma_f32(const float* __restrict__ A, int lda,
                     const float* __restrict__ B, int ldb,
                     const float* __restrict__ bias,
                     float* __restrict__ C, int ldc, int K)
{
    const int lane  = threadIdx.x;          // 0..31, full wave, EXEC all ones
    const int mn    = lane & 15;            // A-row / B-col owned by this lane
    const int khalf = (lane >> 4) * 2;      // 0 or 2: K-pair for this half-wave
    const int row0  = blockIdx.y * 16;
    const int col0  = blockIdx.x * 16;

    const float* Arow = A + (size_t)(row0 + mn) * lda;
    const float* Bcol = B + (size_t)col0 + mn;

    v8f acc = {};
    for (int k = 0; k < K; k += 4) {
        const int kk = k + khalf;
        v2f a = *(const v2f*)(Arow + kk);            // A[m, kk..kk+1] (8B aligned)
        v2f b;
        b.x = Bcol[(size_t)kk * ldb];                // B[kk,   n]
        b.y = Bcol[(size_t)(kk + 1) * ldb];          // B[kk+1, n]
        acc = __builtin_amdgcn_wmma_f32_16x16x4_f32(
                  /*neg_a=*/false, a, /*neg_b=*/false, b,
                  /*c_mod=*/(short)0, acc, /*reuse_a=*/false, /*reuse_b=*/false);
    }

    const float bv    = bias[col0 + mn];             // bias depends on column only
    const int   rbase = row0 + (lane >> 4) * 8;
    #pragma unroll
    for (int i = 0; i < 8; ++i)
        C[(size_t)(rbase + i) * ldc + col0 + mn] = acc[i] + bv;
}

// ---------------------------------------------------------------------------
// out[b, lq, :] = y[b, lq >> 8, :]   (the repeat(v,256) folded through Wo).
// Pure streaming store of 128 MB; y (512 KB) stays L2-resident.
// One 256-thread block per output row, 128-bit per thread, nontemporal stores.
// ---------------------------------------------------------------------------
__global__ __launch_bounds__(256)
void broadcast_rows(const v4f* __restrict__ y, v4f* __restrict__ out)
{
    const int row = blockIdx.x;                  // row = b*16384 + lq
    const int b   = row >> 14;
    const int lq  = row & 16383;
    const int src = (b << 6) + (lq >> 8);        // b*64 + lq/256
    v4f val = y[(size_t)src * 256 + threadIdx.x];
    __builtin_nontemporal_store(val, &out[(size_t)row * 256 + threadIdx.x]);
}

extern "C" void kernel_launch(void* const* d_in, const int* in_sizes, int n_in,
                              void* d_out, int out_size, void* d_ws, size_t ws_size,
                              hipStream_t stream)
{
    // setup_inputs order: x, context, Wq, bq, Wkv, bkv, Wo, bo
    // x/Wq/bq are dead code in the reference — never touched.
    const float* context = (const float*)d_in[1];   // (2, 64, 1024)
    const float* Wkv     = (const float*)d_in[4];   // (1024, 2048)
    const float* bkv     = (const float*)d_in[5];   // (2048,)
    const float* Wo      = (const float*)d_in[6];   // (1024, 1024)
    const float* bo      = (const float*)d_in[7];   // (1024,)
    float*       out     = (float*)d_out;           // (2, 16384, 1024)

    const int D     = 1024;
    const int Rows  = 2 * 64;                       // B*M rows, contiguous

    float* v = (float*)d_ws;                        // (128, 1024) = 512 KB
    float* y = v + (size_t)Rows * D;                // (128, 1024) = 512 KB

    dim3 gblk(D / 16, Rows / 16);                   // 64 x 8 tiles

    // v = context @ Wkv[:, D:2D] + bkv[D:2D]   (V-half of the packed KV proj)
    gemm16_wmma_f32<<<gblk, 32, 0, stream>>>(context, D, Wkv + D, 2 * D,
                                             bkv + D, v, D, D);
    // y = v @ Wo + bo
    gemm16_wmma_f32<<<gblk, 32, 0, stream>>>(v, D, Wo, D, bo, y, D, D);

    // out[b, lq, :] = y[b, lq/256, :]  — the 128 MB streaming write (HBM bound)
    broadcast_rows<<<2 * 16384, 256, 0, stream>>>((const v4f*)y, (v4f*)out);
}